// EmbedLoss_22325240005300
// MI455X (gfx1250) — compile-verified
//
#include <hip/hip_runtime.h>
#include <hip/hip_bf16.h>

// Problem constants (from reference setup_inputs)
#define B_ROWS   8192          // anchors
#define M_CAND   16384         // pos(8192) + neg(8192)
#define DIM      64
#define TOPK     32
#define TILES_N  (M_CAND / 16) // 1024
#define TILES_M  (B_ROWS / 16) // 512

typedef __attribute__((ext_vector_type(2))) float v2f;
typedef __attribute__((ext_vector_type(8))) float v8f;

// ---------------------------------------------------------------------------
// Kernel 1: row L2-normalize (one wave32 per row of 64 floats)
// rows [0,8192)       : anchor -> aN
// rows [8192,16384)   : pos    -> cN[0..8191]
// rows [16384,24576)  : neg    -> cN[8192..16383]
// ---------------------------------------------------------------------------
__global__ __launch_bounds__(256) void normalize_kernel(
    const float* __restrict__ anchor, const float* __restrict__ pos,
    const float* __restrict__ neg, float* __restrict__ aN, float* __restrict__ cN) {
  const int lane = threadIdx.x & 31;
  const int wid  = (blockIdx.x * blockDim.x + threadIdx.x) >> 5;   // global wave id = row id

  const float* src;
  float* dst;
  if (wid < B_ROWS) {
    src = anchor + (size_t)wid * DIM;
    dst = aN + (size_t)wid * DIM;
  } else if (wid < 2 * B_ROWS) {
    src = pos + (size_t)(wid - B_ROWS) * DIM;
    dst = cN + (size_t)(wid - B_ROWS) * DIM;
  } else {
    src = neg + (size_t)(wid - 2 * B_ROWS) * DIM;
    dst = cN + (size_t)(wid - B_ROWS) * DIM;   // appended after pos rows
  }

  float2 v = ((const float2*)src)[lane];       // 2 elems per lane * 32 lanes = 64
  float ss = v.x * v.x + v.y * v.y;
  #pragma unroll
  for (int m = 16; m >= 1; m >>= 1) ss += __shfl_xor(ss, m, 32);
  const float scale = 1.0f / fmaxf(sqrtf(ss), 1e-8f);   // torch eps clamp on norm
  float2 o; o.x = v.x * scale; o.y = v.y * scale;
  ((float2*)dst)[lane] = o;
}

// ---------------------------------------------------------------------------
// Kernel 2: logits = aN (8192x64) * cN^T (64x16384), f32 WMMA 16x16x4.
// One wave32 per 16x16 output tile; K=64 -> 16 chained WMMAs.
// A-layout (32-bit 16x4): lane m = lane&15, k-pair = (lane>>4)*2.
// B (4x16, = cN row-major as K-major per lane): same lane mapping with n = lane&15.
// D (16x16 f32): VGPR v, lanes 0-15 -> row v, lanes 16-31 -> row v+8.
// Uniform control flow => EXEC all ones (WMMA requirement).
// ---------------------------------------------------------------------------
__global__ __launch_bounds__(256) void gemm_wmma_kernel(
    const float* __restrict__ aN, const float* __restrict__ cN,
    float* __restrict__ out) {
  const int lane = threadIdx.x & 31;
  const int wave = threadIdx.x >> 5;
  const int tile = blockIdx.x * 8 + wave;      // 8 waves per block
  const int tN = tile & (TILES_N - 1);
  const int tM = tile >> 10;                   // / TILES_N

  const int mr  = lane & 15;                   // row (A) / col (B) within tile
  const int grp = lane >> 4;                   // selects K pair {0,1} vs {2,3}

  const float* Ap = aN + ((size_t)(tM * 16 + mr) * DIM + grp * 2);
  const float* Bp = cN + ((size_t)(tN * 16 + mr) * DIM + grp * 2);

  v8f acc = {0.f, 0.f, 0.f, 0.f, 0.f, 0.f, 0.f, 0.f};
  #pragma unroll
  for (int k0 = 0; k0 < DIM; k0 += 4) {
    v2f a = *(const v2f*)(Ap + k0);
    v2f b = *(const v2f*)(Bp + k0);
    // (neg_a, A, neg_b, B, c_mod, C, reuse_a, reuse_b)
    acc = __builtin_amdgcn_wmma_f32_16x16x4_f32(false, a, false, b,
                                                (short)0, acc, false, false);
  }

  float* orow = out + ((size_t)(tM * 16 + grp * 8) * M_CAND + (size_t)tN * 16 + mr);
  #pragma unroll
  for (int v = 0; v < 8; ++v) {
    orow[(size_t)v * M_CAND] = acc[v];
  }
}

// ---------------------------------------------------------------------------
// Kernel 3: per-row mask + top-32 select, in place on `out`.
// One 256-thread block per row; each thread holds its 64 elements as
// order-preserving uint keys in REGISTERS (key 0 = "not masked"), so the
// 32-pass MSB radix-select is pure VALU + tiny LDS block reductions.
// ---------------------------------------------------------------------------
__device__ __forceinline__ unsigned orderKey(float f) {
  unsigned u = __float_as_uint(f);
  return u ^ ((u & 0x80000000u) ? 0xFFFFFFFFu : 0x80000000u);  // monotone map
}
__device__ __forceinline__ float invKey(unsigned k) {
  unsigned u = k ^ ((k & 0x80000000u) ? 0x80000000u : 0xFFFFFFFFu);
  return __uint_as_float(u);
}

__device__ __forceinline__ unsigned blockReduceSum(unsigned v, unsigned* sPart,
                                                   unsigned* sBcast) {
  #pragma unroll
  for (int m = 16; m >= 1; m >>= 1) v += (unsigned)__shfl_xor((int)v, m, 32);
  const int lane = threadIdx.x & 31, wave = threadIdx.x >> 5;
  if (lane == 0) sPart[wave] = v;
  __syncthreads();
  if (threadIdx.x == 0) {
    unsigned s = 0;
    #pragma unroll
    for (int w = 0; w < 8; ++w) s += sPart[w];
    *sBcast = s;
  }
  __syncthreads();
  unsigned r = *sBcast;
  __syncthreads();   // safe reuse of sPart/sBcast on next call
  return r;
}

__global__ __launch_bounds__(256) void select_kernel(float* __restrict__ out) {
  __shared__ unsigned sPart[8];
  __shared__ unsigned sBcast;
  __shared__ float sDiag;

  const int t = threadIdx.x;
  const int r = blockIdx.x;
  float* row = out + (size_t)r * M_CAND;

  if (t == 0) sDiag = row[r];   // positive (diagonal) score
  __syncthreads();
  const float diag = sDiag;

  // Load 64 strided elements; key=0 marks "not a semi-hard negative".
  unsigned keys[64];
  unsigned masked = 0;
  #pragma unroll
  for (int i = 0; i < 64; ++i) {
    const float v = row[t + i * 256];
    const bool m = (v < diag);            // strict compare, like reference
    keys[i] = m ? orderKey(v) : 0u;       // finite masked values map to key >= 1
    masked += m ? 1u : 0u;
  }

  const unsigned total = blockReduceSum(masked, sPart, &sBcast);

  unsigned T = 0;   // threshold key: keep masked elems with key >= T
  if (total > TOPK) {
    unsigned prefix = 0, kk = TOPK;
    for (int bit = 31; bit >= 0; --bit) {
      const unsigned want = (prefix >> bit) | 1u;  // bits[31:bit] with bit set
      unsigned c = 0;
      #pragma unroll
      for (int i = 0; i < 64; ++i) c += ((keys[i] >> bit) == want) ? 1u : 0u;
      const unsigned cc = blockReduceSum(c, sPart, &sBcast);
      if (cc >= kk) prefix |= (1u << bit);
      else          kk -= cc;
    }
    T = prefix;   // = key of the TOPK-th largest masked value
  }

  #pragma unroll
  for (int i = 0; i < 64; ++i) {
    const unsigned k = keys[i];
    const bool keep = (k != 0u) && (k >= T);
    row[t + i * 256] = keep ? invKey(k) : 0.0f;   // bit-exact value recovery
  }
}

// ---------------------------------------------------------------------------
extern "C" void kernel_launch(void* const* d_in, const int* in_sizes, int n_in,
                              void* d_out, int out_size, void* d_ws, size_t ws_size,
                              hipStream_t stream) {
  const float* anchor = (const float*)d_in[0];   // [8192, 64]
  const float* pos    = (const float*)d_in[1];   // [8192, 64]
  const float* neg    = (const float*)d_in[2];   // [8192, 64]
  float* out = (float*)d_out;                    // [8192, 16384]

  // Workspace: normalized A (8192*64 f32) then normalized candidates (16384*64 f32) = 6 MB.
  float* aN = (float*)d_ws;
  float* cN = aN + (size_t)B_ROWS * DIM;

  // 24576 rows, 8 waves (rows) per 256-thread block.
  normalize_kernel<<<(B_ROWS * 3) / 8, 256, 0, stream>>>(anchor, pos, neg, aN, cN);

  // 512*1024 tiles, 8 waves (tiles) per block.
  gemm_wmma_kernel<<<(TILES_M * TILES_N) / 8, 256, 0, stream>>>(aN, cN, out);

  // One block per row.
  select_kernel<<<B_ROWS, 256, 0, stream>>>(out);
}